// PopulationCollage_4355096839037
// MI455X (gfx1250) — compile-verified
//
#include <hip/hip_runtime.h>
#include <hip/hip_bf16.h>
#include <math.h>

// PopulationCollage for MI455X (gfx1250, wave32).
// Stage 1: fold per-(pop,patch) affine + colour params into 12-float records.
// Stage 2: gather kernel; per-wave 16-pixel tile, coords via V_WMMA_F32_16X16X4_F32.
// Gather core uses ONE 64-bit base address per (pixel,patch); the 16 bilinear
// taps compile to 8 immediate-offset global_load_b64 (x-adjacent pairs merge).
// Zero-padding handled by separable tap weights with an edge-shift so the base
// is in-bounds by construction. Fully-missed (tile,patch) slots are skipped
// with a wave-uniform vote (no EXEC change -> WMMA-safe).

namespace {
constexpr int S_ = 4;
constexpr int N_ = 48;
constexpr int C_ = 5;
constexpr int H_ = 224;
constexpr int W_ = 224;
constexpr int HW_ = H_ * W_;
constexpr int PSTRIDE = 12;                 // floats per param record
constexpr int TILES_X = W_ / 16;            // 14 tiles per row
constexpr int TILES_PER_S = H_ * TILES_X;   // 3136 (multiple of 8)
constexpr int WAVES_PER_BLOCK = 8;
constexpr int NUM_BLOCKS = (S_ * TILES_PER_S) / WAVES_PER_BLOCK; // 1568
}

typedef __attribute__((ext_vector_type(2))) float v2f;
typedef __attribute__((ext_vector_type(8))) float v8f;

// ---------------------------------------------------------------------------
// Stage 1: per-(s,n) params.
// Record layout: [0]=m00 [1]=m01 [2]=a02 [3]=m10 [4]=m11 [5]=b02 [6..8]=rgb
// px = m00*w + m01*h + a02 ; py = m10*w + m11*h + b02   (source pixel coords)
// ---------------------------------------------------------------------------
__global__ void collage_prep(const float* __restrict__ translation,
                             const float* __restrict__ rotation,
                             const float* __restrict__ scale,
                             const float* __restrict__ squeeze,
                             const float* __restrict__ shear,
                             const float* __restrict__ rgb,
                             float* __restrict__ prm) {
    int i = blockIdx.x * blockDim.x + threadIdx.x;
    if (i >= S_ * N_) return;

    float th = rotation[i] * 6.28318530717958647692f;
    float sc = 0.3f + scale[i] * (2.0f - 0.3f);
    float sq = 0.5f + squeeze[i];
    float sh = (shear[i] * 2.0f - 1.0f) * 0.2f;
    float tx = translation[i * 2 + 0] * 2.0f - 1.0f;
    float ty = translation[i * 2 + 1] * 2.0f - 1.0f;

    float cs = cosf(th), sn = sinf(th);
    float a  = sc * sq;
    float dd = sc / sq;
    float m00 = cs * a;
    float m01 = cs * a * sh - sn * dd;
    float m10 = sn * a;
    float m11 = sn * a * sh + cs * dd;

    // Fold grid_sample (align_corners=False) constants, H_ == W_.
    const float k = 1.0f / (float)W_ - 1.0f;
    float a02 = 0.5f * (float)W_ * (m00 * k + m01 * k + tx + 1.0f) - 0.5f;
    float b02 = 0.5f * (float)H_ * (m10 * k + m11 * k + ty + 1.0f) - 0.5f;

    float* o = prm + (size_t)i * PSTRIDE;
    o[0] = m00; o[1] = m01; o[2] = a02;
    o[3] = m10; o[4] = m11; o[5] = b02;
    o[6] = rgb[i * 3 + 0];
    o[7] = rgb[i * 3 + 1];
    o[8] = rgb[i * 3 + 2];
    o[9] = 0.0f; o[10] = 0.0f; o[11] = 0.0f;
}

// ---------------------------------------------------------------------------
// Stage 2: render. One wave = 16 consecutive output pixels (one row segment).
// Per 8-patch group: one V_WMMA_F32_16X16X4_F32 produces px/py for all
// 16 pixels x 8 patches. Lane L and L+16 share pixel (L&15) and split the
// 8 patches 4/4; partials merged with shfl_xor(16) at the end.
// ---------------------------------------------------------------------------
__global__ __launch_bounds__(256) void collage_main(
        const float* __restrict__ patches,
        const float* __restrict__ prm,
        float* __restrict__ out) {
    __shared__ float sp[N_ * PSTRIDE];

    const int tid   = threadIdx.x;
    const int tile0 = blockIdx.x * WAVES_PER_BLOCK;
    const int s     = tile0 / TILES_PER_S;   // block never straddles s

    for (int i = tid; i < N_ * PSTRIDE; i += 256)
        sp[i] = prm[s * N_ * PSTRIDE + i];
    __syncthreads();

    const int wave = tid >> 5;
    const int lane = tid & 31;
    const int tile = tile0 + wave;
    const int rem  = tile - s * TILES_PER_S;
    const int h    = rem / TILES_X;
    const int w0   = (rem % TILES_X) * 16;

    const int half   = lane >> 4;   // 0: K=0,1 / rows 0-7 ; 1: K=2,3 / rows 8-15
    const int r2     = lane & 15;
    const int pg_row = r2 >> 1;     // A-matrix row -> patch-in-group
    const int axis   = r2 & 1;      // 0 -> px row, 1 -> py row

    // B matrix (4x16), loop invariant: columns are pixels (w, h, 1, 0).
    v2f Bm;
    Bm.x = half ? 1.0f : (float)(w0 + r2);   // K=0 (w) | K=2 (1)
    Bm.y = half ? 0.0f : (float)h;           // K=1 (h) | K=3 (0)

    const float* pbaseS = patches + (size_t)s * N_ * C_ * HW_;

    float accr = 0.0f, accg = 0.0f, accb = 0.0f;

#pragma unroll 2
    for (int g = 0; g < N_ / 8; ++g) {
        // A matrix row (patch = g*8 + pg_row, axis) coefficients.
        const float* q = sp + (g * 8 + pg_row) * PSTRIDE + axis * 3;
        float c0 = q[0], c1 = q[1], c2 = q[2];
        v2f Am;
        Am.x = half ? c2 : c0;
        Am.y = half ? 0.0f : c1;

        v8f cz = {0.f, 0.f, 0.f, 0.f, 0.f, 0.f, 0.f, 0.f};
        v8f dm = __builtin_amdgcn_wmma_f32_16x16x4_f32(
            /*neg_a=*/false, Am, /*neg_b=*/false, Bm,
            /*c_mod=*/(short)0, cz, /*reuse_a=*/false, /*reuse_b=*/false);

        const int patch0 = g * 8 + half * 4;  // this half-wave's 4 patches
#pragma unroll
        for (int slot = 0; slot < 4; ++slot) {
            const int patch = patch0 + slot;
            float px = dm[2 * slot + 0];
            float py = dm[2 * slot + 1];

            float x0f = floorf(px), y0f = floorf(py);
            float wx1 = px - x0f, wy1 = py - y0f;
            float wx0 = 1.0f - wx1, wy0 = 1.0f - wy1;
            int x0 = (int)x0f, y0 = (int)y0f;

            // Separable tap weights with zero-pad masking; shift the valid
            // right/bottom tap onto tap0 at the -1 edge so the clamped base
            // address is correct and >= 0 by construction.
            float vx0 = ((unsigned)x0       < (unsigned)W_) ? wx0 : 0.0f;
            float vx1 = ((unsigned)(x0 + 1) < (unsigned)W_) ? wx1 : 0.0f;
            vx0 = (x0 == -1) ? vx1 : vx0;
            vx1 = (x0 == -1) ? 0.0f : vx1;
            int xc = min(max(x0, 0), W_ - 1);

            float vy0 = ((unsigned)y0       < (unsigned)H_) ? wy0 : 0.0f;
            float vy1 = ((unsigned)(y0 + 1) < (unsigned)H_) ? wy1 : 0.0f;
            vy0 = (y0 == -1) ? vy1 : vy0;
            vy1 = (y0 == -1) ? 0.0f : vy1;
            int yc = min(max(y0, 0), H_ - 1);

            // Wave-uniform early-out: if every lane's sample for this slot
            // lies fully in the zero-padding region, skip the 8 gathers.
            // Uniform branch -> EXEC untouched (WMMA-safe).
            bool hit = ((vx0 + vx1) * (vy0 + vy1)) > 0.0f;
            if (__any(hit)) {
                float w00 = vy0 * vx0, w01 = vy0 * vx1;
                float w10 = vy1 * vx0, w11 = vy1 * vx1;

                // Single base address; 16 taps -> 8 global_load_b64 with
                // immediate offsets. Masked taps may read up to one row past
                // channel c, which stays inside this patch (channel c+1 <= 4
                // always exists).
                const float* p = pbaseS + (size_t)patch * C_ * HW_
                               + yc * W_ + xc;
                float vch[4];
#pragma unroll
                for (int c = 0; c < 4; ++c) {
                    const float* pc = p + c * HW_;
                    vch[c] = pc[0]  * w00 + pc[1]      * w01
                           + pc[W_] * w10 + pc[W_ + 1] * w11;
                }

                const float* col = sp + patch * PSTRIDE + 6;
                float va = vch[3];
                accr += vch[0] * col[0] * va;
                accg += vch[1] * col[1] * va;
                accb += vch[2] * col[2] * va;
            }
        }
    }

    // Merge the two patch-halves that share each pixel (wave32 xor-16).
    accr += __shfl_xor(accr, 16, 32);
    accg += __shfl_xor(accg, 16, 32);
    accb += __shfl_xor(accb, 16, 32);

    if (half == 0) {
        const int w = w0 + r2;
        float* o = out + ((size_t)(s * H_ + h) * W_ + w) * 3;
        o[0] = fminf(fmaxf(accr, 0.0f), 1.0f);
        o[1] = fminf(fmaxf(accg, 0.0f), 1.0f);
        o[2] = fminf(fmaxf(accb, 0.0f), 1.0f);
    }
}

extern "C" void kernel_launch(void* const* d_in, const int* in_sizes, int n_in,
                              void* d_out, int out_size, void* d_ws, size_t ws_size,
                              hipStream_t stream) {
    const float* patches     = (const float*)d_in[0];
    const float* translation = (const float*)d_in[1];
    const float* rotation    = (const float*)d_in[2];
    const float* scale       = (const float*)d_in[3];
    const float* squeeze     = (const float*)d_in[4];
    const float* shear       = (const float*)d_in[5];
    const float* rgb         = (const float*)d_in[6];
    // d_in[7] = order: multiplies channel 4 only, which never reaches output.

    float* prm = (float*)d_ws;   // S*N*12 floats = 9216 B
    float* out = (float*)d_out;  // [S,H,W,3] f32

    collage_prep<<<1, S_ * N_, 0, stream>>>(translation, rotation, scale,
                                            squeeze, shear, rgb, prm);
    collage_main<<<NUM_BLOCKS, 256, 0, stream>>>(patches, prm, out);
}